// Resample_26362509262915
// MI455X (gfx1250) — compile-verified
//
#include <hip/hip_runtime.h>

// Rational resampler up=2 / down=3, 128-tap FIR, f32 in/out.
// out[q] = 2 * sum_{s=0..63} c[q&1][s] * x[base(q)+s],  base(q) = (3q-63)>>1
//   c0[s] = h[127-2s]  (even outputs),  c1[s] = h[126-2s] (odd outputs)
//
// Mapped onto V_WMMA_F32_16X16X4_F32 via a Toeplitz/im2col factorization:
//   D[m][n] = sum_{j=0..87} A[m][j] * B[j][n]
//   A[m][j] = x[tileBase + 24*m + j]          (16 row windows, step 24)
//   B[j][n] = shifted/zero-padded phase filter (fixed 88x16, precomputed)
// 22 chained K=4 WMMAs accumulate the K=88 contraction in f32.

typedef float v2f __attribute__((ext_vector_type(2)));
typedef float v8f __attribute__((ext_vector_type(8)));

#define KDIM          88          // padded contraction length (multiple of 4)
#define NCHUNK        (KDIM / 4)  // 22 WMMA steps per tile
#define WAVES_PER_WG  8
#define WG_THREADS    (WAVES_PER_WG * 32)
#define OUTS_PER_TILE 256         // 16x16 outputs per wave
#define X_PER_TILE    384         // input advance per 256 outputs (256*3/2)
#define LDS_X         (X_PER_TILE * (WAVES_PER_WG - 1) + 448) // 3136 floats
#define BPACK_ELEMS   (NCHUNK * 64)                           // 1408 floats

// ---------------------------------------------------------------------------
// Prep kernel: build the 88x16 B matrix from h, already laid out in the WMMA
// B-operand lane order: for chunk kk, lane L holds rows {k0,k0+1}, k0=(L<16?0:2),
// column n = L&15.  bpack[(kk*32 + L)*2 + e] = B[4*kk + k0 + e][n], scaled by 2.
// ---------------------------------------------------------------------------
__global__ void resample_build_bpack(const float* __restrict__ h,
                                     float* __restrict__ bpack) {
    int idx = blockIdx.x * blockDim.x + threadIdx.x;
    if (idx >= BPACK_ELEMS) return;
    int kk  = idx >> 6;        // chunk
    int rem = idx & 63;        // L*2 + e
    int L   = rem >> 1;
    int e   = rem & 1;
    int n   = L & 15;
    int k0  = (L < 16) ? 0 : 2;
    int j   = 4 * kk + k0 + e; // row in [0, 87]
    float v = 0.0f;
    if (n & 1) {
        int s = j - (3 * n + 1) / 2;            // odd phase shift
        if (s >= 0 && s < 64) v = 2.0f * h[126 - 2 * s];
    } else {
        int s = j - (3 * n) / 2;                // even phase shift
        if (s >= 0 && s < 64) v = 2.0f * h[127 - 2 * s];
    }
    bpack[idx] = v;
}

// ---------------------------------------------------------------------------
// Main kernel: 8 waves per workgroup, one 16x16 output tile per wave.
// x staged once through LDS with zero padding at the signal edges.
// ---------------------------------------------------------------------------
__global__ __launch_bounds__(WG_THREADS)
void resample_wmma_kernel(const float* __restrict__ x,
                          const float* __restrict__ bpack,
                          float* __restrict__ out,
                          int N, int out_len) {
    __shared__ float lds_x[LDS_X];
    __shared__ float lds_b[BPACK_ELEMS];

    const int tid = threadIdx.x;
    const int T0  = blockIdx.x * WAVES_PER_WG;          // first tile of this WG
    const int B0  = X_PER_TILE * T0 - 32;               // base(256*T0)

    // Stage x (zero-padded out-of-range) and the B matrix into LDS.
    for (int t = tid; t < LDS_X; t += WG_THREADS) {
        int gi = B0 + t;
        lds_x[t] = (gi >= 0 && gi < N) ? x[gi] : 0.0f;
    }
    for (int t = tid; t < BPACK_ELEMS; t += WG_THREADS)
        lds_b[t] = bpack[t];
    __syncthreads();

    const int wave = tid >> 5;
    const int lane = tid & 31;
    const int m    = lane & 15;               // A-row / D-column index
    const int k0   = (lane < 16) ? 0 : 2;     // K sub-pair per ISA A layout

    // A operand: lane reads x[tile + 24*m + 4*kk + k0 .. +1] (8B aligned).
    const int abase = X_PER_TILE * wave + 24 * m + k0;
    const int bbase = lane * 2;

    v8f c = {};
#pragma unroll
    for (int kk = 0; kk < NCHUNK; ++kk) {
        v2f a = *(const v2f*)(&lds_x[abase + 4 * kk]);
        v2f b = *(const v2f*)(&lds_b[kk * 64 + bbase]);
        // (neg_a, A, neg_b, B, c_mod, C, reuse_a, reuse_b)
        c = __builtin_amdgcn_wmma_f32_16x16x4_f32(
                false, a, false, b, (short)0, c, false, false);
    }

    // D layout: VGPR r, lane L -> M = r + (L<16 ? 0 : 8), N = L&15.
    // Output index q = q0 + 16*M + N.
    const int q0 = (T0 + wave) * OUTS_PER_TILE;
    const int mrow = (lane < 16) ? 0 : 8;
#pragma unroll
    for (int r = 0; r < 8; ++r) {
        int q = q0 + 16 * (r + mrow) + m;
        if (q < out_len) out[q] = c[r];
    }
}

// ---------------------------------------------------------------------------
extern "C" void kernel_launch(void* const* d_in, const int* in_sizes, int n_in,
                              void* d_out, int out_size, void* d_ws, size_t ws_size,
                              hipStream_t stream) {
    (void)n_in; (void)ws_size;
    const float* x = (const float*)d_in[0];
    const float* h = (const float*)d_in[1];
    // d_in[2] (up=2) and d_in[3] (down=3) are compile-time constants here.
    float* out   = (float*)d_out;
    float* bpack = (float*)d_ws;               // 1408 floats of scratch
    int N = in_sizes[0];

    resample_build_bpack<<<(BPACK_ELEMS + 255) / 256, 256, 0, stream>>>(h, bpack);

    int tiles  = (out_size + OUTS_PER_TILE - 1) / OUTS_PER_TILE;
    int blocks = (tiles + WAVES_PER_WG - 1) / WAVES_PER_WG;
    resample_wmma_kernel<<<blocks, WG_THREADS, 0, stream>>>(x, bpack, out, N, out_size);
}